// TopKSAE_23055384445818
// MI455X (gfx1250) — compile-verified
//
#include <hip/hip_runtime.h>

#define D_VIT  768
#define D_SAE  12288
#define TOPK   32
#define ROWS_PER_BLOCK 16
#define WAVES  16
#define THREADS (WAVES * 32)
#define CHUNK_COLS (WAVES * 16)          // 256
#define NCHUNKS (D_SAE / CHUNK_COLS)     // 48
#define COLS_PER_LANE (D_VIT / 32)       // 24

typedef __attribute__((ext_vector_type(2))) float v2f;
typedef __attribute__((ext_vector_type(8))) float v8f;

// ---- wave32 bitonic helpers: (val,idx) pairs, descending by value,
// ---- ties broken toward lower index (matches lax.top_k preference).

__device__ __forceinline__ bool pair_less(float pv, unsigned pi, float v, unsigned i) {
    // strict total order: "partner < mine"
    return (pv < v) || (pv == v && pi > i);
}

__device__ __forceinline__ void sort_desc32(float& v, unsigned& id, int lane) {
#pragma unroll
    for (int k = 2; k <= 32; k <<= 1) {
#pragma unroll
        for (int j = k >> 1; j >= 1; j >>= 1) {
            float    pv = __shfl_xor(v, j, 32);
            unsigned pi = (unsigned)__shfl_xor((int)id, j, 32);
            bool dirAsc    = (lane & k) != 0;                 // final pass: descending
            bool keepSmall = (((lane & j) == 0) == dirAsc);
            if (pair_less(pv, pi, v, id) == keepSmall) { v = pv; id = pi; }
        }
    }
}

__device__ __forceinline__ void bitonic_merge_desc32(float& v, unsigned& id, int lane) {
#pragma unroll
    for (int j = 16; j >= 1; j >>= 1) {
        float    pv = __shfl_xor(v, j, 32);
        unsigned pi = (unsigned)__shfl_xor((int)id, j, 32);
        bool keepSmall = ((lane & j) != 0);                   // lower lane keeps larger
        if (pair_less(pv, pi, v, id) == keepSmall) { v = pv; id = pi; }
    }
}

__global__ __launch_bounds__(THREADS, 1)
void topk_sae_fused(const float* __restrict__ x,
                    const float* __restrict__ W_enc,
                    const float* __restrict__ W_dec,
                    const float* __restrict__ b_dec,
                    float* __restrict__ out) {
    __shared__ float xs[D_VIT * ROWS_PER_BLOCK];          // transposed [k][m], 48KB
    __shared__ float acts[ROWS_PER_BLOCK * CHUNK_COLS];   // [m][col], 16KB

    const int tid  = threadIdx.x;
    const int lane = tid & 31;
    const int wave = tid >> 5;
    const int rowBase = blockIdx.x * ROWS_PER_BLOCK;

    // ---- Stage 0: load x tile (centered by b_dec), store transposed in LDS
    for (int i = tid; i < ROWS_PER_BLOCK * D_VIT; i += THREADS) {
        int m = i / D_VIT;
        int k = i - m * D_VIT;
        float v = x[(rowBase + m) * D_VIT + k] - b_dec[k];
        xs[k * ROWS_PER_BLOCK + m] = v;
    }
    __syncthreads();

    // running per-row top-32, sorted descending across lanes of each wave
    float    tv   = -1.0f;   // relu output >= 0, so -1 guarantees first merge
    unsigned ti   = 0u;
    float    cmin = -1.0f;   // cached copy of lane-31 value of tv (k-th largest)

    const int kb = (lane >> 4) << 1;   // 0 (lanes 0-15) or 2 (lanes 16-31)
    const int nn = lane & 15;          // M for A-frag, N for B/C-frags

    for (int ch = 0; ch < NCHUNKS; ++ch) {
        // ---- Stage 1: each wave computes one 16x16 acts tile via fp32 WMMA
        const int colBase = ch * CHUNK_COLS + wave * 16;
        const float* __restrict__ wcol = W_enc + colBase + nn;

        v8f c = {};
#pragma unroll 8
        for (int k = 0; k < D_VIT; k += 4) {
            v2f a, b;
            a.x = xs[(k + kb)     * ROWS_PER_BLOCK + nn];
            a.y = xs[(k + kb + 1) * ROWS_PER_BLOCK + nn];
            b.x = wcol[(k + kb)     * D_SAE];
            b.y = wcol[(k + kb + 1) * D_SAE];
            c = __builtin_amdgcn_wmma_f32_16x16x4_f32(
                    false, a, false, b, (short)0, c, false, false);
        }

        // relu + spill C tile to LDS (C layout: VGPR r -> M = r + 8*(lane>>4), N = nn)
        const int mBase = (lane >> 4) << 3;
#pragma unroll
        for (int r = 0; r < 8; ++r) {
            acts[(mBase + r) * CHUNK_COLS + wave * 16 + nn] = fmaxf(c[r], 0.0f);
        }
        __syncthreads();

        // ---- Stage 2: wave w folds chunk of row w into its running top-32
        {
            const int row = wave;
#pragma unroll
            for (int g = 0; g < CHUNK_COLS / 32; ++g) {
                float    v  = acts[row * CHUNK_COLS + g * 32 + lane];
                unsigned id = (unsigned)(ch * CHUNK_COLS + g * 32 + lane);

                // wave-uniform early reject: any lane beating current k-th value?
                if (__ballot(v > cmin)) {
                    sort_desc32(v, id, lane);
                    // reverse batch (descending -> ascending), pairwise max
                    float    rv = __shfl_xor(v, 31, 32);
                    unsigned ri = (unsigned)__shfl_xor((int)id, 31, 32);
                    bool pGreater = (rv > tv) || (rv == tv && ri < ti);
                    if (pGreater) { tv = rv; ti = ri; }
                    // result is bitonic; 5-stage merge back to descending
                    bitonic_merge_desc32(tv, ti, lane);
                    cmin = __shfl(tv, 31, 32);   // refresh threshold (rare path)
                }
            }
        }
        __syncthreads();
    }

    // ---- Stage 3: sparse decode. wave w writes row rowBase + w.
    {
        const int row = rowBase + wave;
        float acc[COLS_PER_LANE];
#pragma unroll
        for (int cidx = 0; cidx < COLS_PER_LANE; ++cidx) acc[cidx] = 0.0f;

        for (int j = 0; j < TOPK; ++j) {
            float    vj = __shfl(tv, j, 32);
            unsigned ij = (unsigned)__shfl((int)ti, j, 32);
            if (vj > 0.0f) {   // wave-uniform
                const float* __restrict__ wd = W_dec + ij * D_VIT + lane;
#pragma unroll
                for (int cidx = 0; cidx < COLS_PER_LANE; ++cidx)
                    acc[cidx] = fmaf(vj, wd[cidx * 32], acc[cidx]);
            }
        }

        float* __restrict__ op = out + row * D_VIT + lane;
#pragma unroll
        for (int cidx = 0; cidx < COLS_PER_LANE; ++cidx)
            op[cidx * 32] = acc[cidx] + b_dec[lane + cidx * 32];
    }
}

extern "C" void kernel_launch(void* const* d_in, const int* in_sizes, int n_in,
                              void* d_out, int out_size, void* d_ws, size_t ws_size,
                              hipStream_t stream) {
    (void)n_in; (void)d_ws; (void)ws_size; (void)out_size;
    const float* x     = (const float*)d_in[0];
    const float* W_enc = (const float*)d_in[1];
    const float* W_dec = (const float*)d_in[2];
    const float* b_dec = (const float*)d_in[3];
    float* out = (float*)d_out;

    const int n = in_sizes[0] / D_VIT;           // 16384 rows
    dim3 grid(n / ROWS_PER_BLOCK);               // 1024 blocks
    topk_sae_fused<<<grid, THREADS, 0, stream>>>(x, W_enc, W_dec, b_dec, out);
}